// WaveletEncoding_29128468202080
// MI455X (gfx1250) — compile-verified
//
#include <hip/hip_runtime.h>

// ---------------------------------------------------------------------------
// db4 3D DWT, fp32. Two passes:
//   K1: per (b,c,t) slice, fused W+H DWT via LDS tile; tile filled with
//       GLOBAL_LOAD_ASYNC_TO_LDS_B32 (ASYNCcnt path), W-stage runs on
//       V_WMMA_F32_16X16X4_F32 (im2col matmul), H-stage on VALU FMAs.
//       Writes 4 subbands to d_ws: [B*C*T, 4, 128, 128]  (96 MB, L2-resident)
//   K2: T-stage streaming kernel, writes final [B, 8*C, 16, 128, 128].
// Reflect-3 padding folded in; both stages use y[j] = sum_t Frev[t]*xpad[2j+t]
// with Frev = reversed dec filter (derived from jax conv with flipped kernel).
// ---------------------------------------------------------------------------

typedef __attribute__((ext_vector_type(2))) float v2f;
typedef __attribute__((ext_vector_type(8))) float v8f;

// Reversed db4 decomposition filters (Frev[t] = dec[7-t])
constexpr float RLO[8] = {
     0.23037781330885523f,  0.7148465705525415f,   0.6308807679295904f,
    -0.02798376941698385f, -0.18703481171888114f,  0.030841381835986965f,
     0.032883011666982945f, -0.010597401784997278f };
constexpr float RHI[8] = {
    -0.010597401784997278f, -0.032883011666982945f, 0.030841381835986965f,
     0.18703481171888114f,  -0.02798376941698385f, -0.6308807679295904f,
     0.7148465705525415f,   -0.23037781330885523f };

__device__ __forceinline__ int reflect_idx(int i, int L) {
    return i < 0 ? -i : (i >= L ? 2 * L - 2 - i : i);
}

// Direct global->LDS async copy, one dword per lane.
// VDST holds the LDS byte offset from the LDS base.
// Low 32 bits of a generic __shared__ pointer are exactly that offset
// (flat->LDS mapping truncates to addr[31:0]).
__device__ __forceinline__ void async_load_lds_b32(unsigned lds_off,
                                                   const float* gaddr) {
    asm volatile("global_load_async_to_lds_b32 %0, %1, off"
                 :: "v"(lds_off), "v"(gaddr)
                 : "memory");
}
__device__ __forceinline__ void wait_asynccnt0() {
    asm volatile("s_wait_asynccnt 0x0" ::: "memory");
}

// ---------------------------------------------------------------------------
// Kernel 1: fused W+H DWT on one t-slice.
// grid = (B*C*T = 384, H2/16 = 8, W2/32 = 4), block = 256 (8 wave32)
// ---------------------------------------------------------------------------
__global__ __launch_bounds__(256)
void dwt_hw_wmma_kernel(const float* __restrict__ x, float* __restrict__ y2)
{
    constexpr int W = 256, H = 256;
    __shared__ float sIn[38 * 72];        // reflect-padded input tile 38x70
    __shared__ float sAW[2 * 38 * 33];    // W-stage result: [lo/hi][38 rows][32]

    const int bct = blockIdx.x;           // (b*3+c)*32 + t
    const int h0  = blockIdx.y * 16;      // output h2 base
    const int w0  = blockIdx.z * 32;      // output w2 base
    const int tid = threadIdx.x;
    const size_t ibase = (size_t)bct * H * W;

    // ---- fill 38x70 tile with reflect padding: async global->LDS ----
    for (int i = tid; i < 38 * 70; i += 256) {
        int r = i / 70, c = i - r * 70;
        int gh = reflect_idx(2 * h0 - 3 + r, H);
        int gw = reflect_idx(2 * w0 - 3 + c, W);
        unsigned lds_off = (unsigned)(uintptr_t)(&sIn[r * 72 + c]);
        async_load_lds_b32(lds_off, x + ibase + (size_t)gh * W + gw);
    }
    wait_asynccnt0();      // drain this wave's async copies
    __syncthreads();       // make all waves' tile portions visible

    // ---- W-stage: im2col matmul on V_WMMA_F32_16X16X4_F32 ----
    // D(16x16) = A(16x4) x B(4x16) + C, chained twice for the 8 taps.
    // A rows: M=0 -> RLO taps, M=1 -> RHI taps, rows 2..15 zero.
    // f32 operand striping (mirrors documented A/D layouts):
    //   A: lanes 0-15 hold M=lane K={0,1} in vgpr{0,1}; lanes 16-31 K={2,3}
    //   B: lane half h, vgpr v holds row K=2h+v, column N=lane&15
    //   D: vgpr v, lanes 0-15 -> M=v, N=lane  (v=0 -> lo, v=1 -> hi)
    const int lane = tid & 31;
    const int wave = tid >> 5;
    const int mrow = lane & 15;
    const int half = lane >> 4;

    float f0 = 0.f, f1 = 0.f, f4 = 0.f, f5 = 0.f;
    if (mrow == 0) {
        f0 = half ? RLO[2] : RLO[0];  f1 = half ? RLO[3] : RLO[1];
        f4 = half ? RLO[6] : RLO[4];  f5 = half ? RLO[7] : RLO[5];
    } else if (mrow == 1) {
        f0 = half ? RHI[2] : RHI[0];  f1 = half ? RHI[3] : RHI[1];
        f4 = half ? RHI[6] : RHI[4];  f5 = half ? RHI[7] : RHI[5];
    }
    const v2f A0 = { f0, f1 };   // taps 0..3
    const v2f A1 = { f4, f5 };   // taps 4..7

    for (int r = wave; r < 38; r += 8) {            // wave-uniform: EXEC all 1s
        const float* row = &sIn[r * 72];
#pragma unroll
        for (int jb = 0; jb < 2; ++jb) {            // two 16-wide output groups
            const int cb = 2 * (jb * 16 + mrow) + 2 * half;
            v2f B0 = { row[cb],     row[cb + 1] };  // taps 0..3 at col 2j
            v2f B1 = { row[cb + 4], row[cb + 5] };  // taps 4..7
            v8f d = {0.f,0.f,0.f,0.f,0.f,0.f,0.f,0.f};
            d = __builtin_amdgcn_wmma_f32_16x16x4_f32(false, A0, false, B0,
                                                      (short)0, d, false, false);
            d = __builtin_amdgcn_wmma_f32_16x16x4_f32(false, A1, false, B1,
                                                      (short)0, d, false, false);
            if (lane < 16) {
                const int j = jb * 16 + lane;
                sAW[(0 * 38 + r) * 33 + j] = d[0];  // lo_w (D row M=0)
                sAW[(1 * 38 + r) * 33 + j] = d[1];  // hi_w (D row M=1)
            }
        }
    }
    __syncthreads();

    // ---- H-stage (VALU FMA) + store 4 subbands: sb2 = h_sel*2 + w_sel ----
#pragma unroll
    for (int hs = 0; hs < 2; ++hs) {
#pragma unroll
        for (int ws = 0; ws < 2; ++ws) {
            for (int i = tid; i < 512; i += 256) {  // 16 h2 x 32 w2
                const int m = i >> 5, j = i & 31;
                float acc = 0.f;
#pragma unroll
                for (int t = 0; t < 8; ++t) {
                    const float f = hs ? RHI[t] : RLO[t];   // folds to literal
                    acc += f * sAW[(ws * 38 + 2 * m + t) * 33 + j];
                }
                const int sb2 = hs * 2 + ws;
                const size_t o = (((size_t)bct * 4 + sb2) * 128 + (h0 + m)) * 128
                               + (w0 + j);
                y2[o] = acc;
            }
        }
    }
}

// ---------------------------------------------------------------------------
// Kernel 2: T-stage. One thread per (b,c,sb2,h2,w2) column; 32 t-loads into
// registers (reflect indices constant-folded by full unroll), 16 lo/hi pairs.
// grid = 786432/256 = 3072 blocks (exact), block = 256.
// ---------------------------------------------------------------------------
__global__ __launch_bounds__(256)
void dwt_t_kernel(const float* __restrict__ y2, float* __restrict__ out)
{
    const int n   = blockIdx.x * 256 + threadIdx.x;  // < 12*4*128*128
    const int w2  = n & 127;
    const int h2  = (n >> 7) & 127;
    const int sb2 = (n >> 14) & 3;
    const int bc  = n >> 16;                         // 0..11
    const int b   = bc / 3, c = bc - 3 * b;

    const size_t colbase = ((size_t)bc * 32 * 4 + sb2) * 16384
                         + (size_t)h2 * 128 + w2;
    float xv[32];
#pragma unroll
    for (int t = 0; t < 32; ++t)
        xv[t] = y2[colbase + (size_t)t * (4 * 16384)];

    const int ch_lo = sb2 * 3 + c;         // s = 0*4 + sb2
    const int ch_hi = (4 + sb2) * 3 + c;   // s = 1*4 + sb2
    const size_t obl = ((size_t)(b * 24 + ch_lo) * 16) * 16384
                     + (size_t)h2 * 128 + w2;
    const size_t obh = ((size_t)(b * 24 + ch_hi) * 16) * 16384
                     + (size_t)h2 * 128 + w2;

#pragma unroll
    for (int j = 0; j < 16; ++j) {
        float lo = 0.f, hi = 0.f;
#pragma unroll
        for (int t = 0; t < 8; ++t) {
            const int mm = 2 * j - 3 + t;                       // [-3, 34]
            const int rt = mm < 0 ? -mm : (mm > 31 ? 62 - mm : mm);
            const float v = xv[rt];                             // reg (const idx)
            lo += RLO[t] * v;
            hi += RHI[t] * v;
        }
        out[obl + (size_t)j * 16384] = lo;
        out[obh + (size_t)j * 16384] = hi;
    }
}

// ---------------------------------------------------------------------------
extern "C" void kernel_launch(void* const* d_in, const int* in_sizes, int n_in,
                              void* d_out, int out_size, void* d_ws, size_t ws_size,
                              hipStream_t stream)
{
    (void)in_sizes; (void)n_in; (void)out_size; (void)ws_size;
    const float* x  = (const float*)d_in[0];   // [4,3,32,256,256] f32
    // d_in[1] (filters) matches the hardcoded db4 constants; not re-read.
    float* y2  = (float*)d_ws;                 // needs 4*3*32*4*128*128*4 = 96 MB
    float* out = (float*)d_out;                // [4,24,16,128,128] f32

    dwt_hw_wmma_kernel<<<dim3(384, 8, 4), 256, 0, stream>>>(x, y2);
    dwt_t_kernel<<<dim3(3072), 256, 0, stream>>>(y2, out);
}